// Loss_ssi_basic_83227876262126
// MI455X (gfx1250) — compile-verified
//
#include <hip/hip_runtime.h>

typedef unsigned int u32;
typedef unsigned char u8;
typedef __attribute__((ext_vector_type(2))) float v2f;
typedef __attribute__((ext_vector_type(8))) float v8f;

#define NIMG 64
#define HD   518
#define WD   518
#define NPIX (HD * WD)            // 268324
#define TOT  (NIMG * NPIX)        // 17172736
#define H1 259
#define W1 259
#define C1 (H1 * W1)              // 67081
#define C2 (129 * 129)            // 16641
#define C3 (64 * 64)              // 4096
#define EPSV 1e-8f

// ---- order-preserving float <-> u32 key ----
__device__ __forceinline__ unsigned fkey(float f) {
  unsigned u = __float_as_uint(f);
  return (u & 0x80000000u) ? ~u : (u | 0x80000000u);
}
__device__ __forceinline__ float fval(unsigned k) {
  unsigned u = (k & 0x80000000u) ? (k & 0x7fffffffu) : ~k;
  return __uint_as_float(u);
}

// ---- block (256 thr) sum reduction; tail done with V_WMMA_F32_16X16X4_F32 ----
// 256 partials -> 64 values packed 2/lane in wave0 (= 16x4 f32 A layout),
// A x ones(4x16) gives row sums in D; total = lane0 + lane16 of sum(D regs).
__device__ __forceinline__ float block_sum(float v, float* lds) {
  const int tid = threadIdx.x;
  lds[tid] = v;
  __syncthreads();
  if (tid < 32) {
    v2f a;
    a[0] = lds[tid] + lds[tid + 64] + lds[tid + 128] + lds[tid + 192];
    a[1] = lds[tid + 32] + lds[tid + 96] + lds[tid + 160] + lds[tid + 224];
    v2f b; b[0] = 1.0f; b[1] = 1.0f;
    v8f c = {0.f, 0.f, 0.f, 0.f, 0.f, 0.f, 0.f, 0.f};
    v8f d = __builtin_amdgcn_wmma_f32_16x16x4_f32(false, a, false, b,
                                                  (short)0, c, false, false);
    float t = d[0] + d[1] + d[2] + d[3] + d[4] + d[5] + d[6] + d[7];
    float tot = __shfl(t, 0, 32) + __shfl(t, 16, 32);
    if (tid == 0) lds[0] = tot;
  }
  __syncthreads();
  float r = lds[0];
  __syncthreads();
  return r;
}

// ---- zero scratch (hists + cnt) ----
__global__ void k_zero(u32* __restrict__ p, int n) {
  for (int i = blockIdx.x * blockDim.x + threadIdx.x; i < n;
       i += gridDim.x * blockDim.x)
    p[i] = 0;
}

// ---- masked radix histogram pass (level 1/2/3), LDS 2048-bin histograms ----
__global__ __launch_bounds__(256) void k_hist(
    const float* __restrict__ pred, const float* __restrict__ yv,
    const u8* __restrict__ mask, u32* __restrict__ hist, u32* __restrict__ cnt,
    const u32* __restrict__ b1, const u32* __restrict__ b2, int level,
    int nbins) {
  __shared__ u32 h[2048];
  __shared__ u32 rsum[256];
  const int tid = threadIdx.x;
  for (int i = tid; i < nbins; i += 256) h[i] = 0;
  const int img = blockIdx.y;
  const int t = blockIdx.z;
  const float* src = t ? yv : pred;
  unsigned topsel = 0;
  if (level == 2) topsel = b1[t * NIMG + img];
  else if (level == 3) topsel = (b1[t * NIMG + img] << 11) | b2[t * NIMG + img];
  __syncthreads();
  const int base = img * NPIX;
  for (int i = blockIdx.x * 256 + tid; i < NPIX; i += gridDim.x * 256) {
    if (mask[base + i]) {
      unsigned key = fkey(src[base + i]);
      if (level == 1) {
        atomicAdd(&h[key >> 21], 1u);
      } else if (level == 2) {
        if ((key >> 21) == topsel) atomicAdd(&h[(key >> 10) & 2047u], 1u);
      } else {
        if ((key >> 10) == topsel) atomicAdd(&h[key & 1023u], 1u);
      }
    }
  }
  __syncthreads();
  unsigned tot = 0;
  u32* gh = hist + (size_t)(t * NIMG + img) * nbins;
  for (int i = tid; i < nbins; i += 256) {
    u32 v = h[i];
    if (v) atomicAdd(&gh[i], v);
    tot += v;
  }
  if (level == 1 && t == 0) {  // mask count == sum of all bins
    rsum[tid] = tot;
    __syncthreads();
    for (int s = 128; s > 0; s >>= 1) {
      if (tid < s) rsum[tid] += rsum[tid + s];
      __syncthreads();
    }
    if (tid == 0 && rsum[0]) atomicAdd(&cnt[img], rsum[0]);
  }
}

// ---- pick bucket containing rank k; level 3 also reconstructs the median ----
__global__ __launch_bounds__(256) void k_select(
    const u32* __restrict__ hist, const u32* __restrict__ cnt,
    const u32* __restrict__ rin, u32* __restrict__ bout, u32* __restrict__ rout,
    const u32* __restrict__ b1, const u32* __restrict__ b2,
    float* __restrict__ med, int level, int nbins) {
  const int job = blockIdx.x;           // 0..127 = t*64+img
  const int t = job / NIMG, img = job % NIMG;
  const unsigned c = cnt[img];
  unsigned k = (level == 1) ? (c ? ((c - 1u) >> 1) : 0u) : rin[t * NIMG + img];
  const u32* h = hist + (size_t)job * nbins;
  const int tid = threadIdx.x;
  const int per = nbins / 256;
  __shared__ u32 ps[256];
  unsigned s = 0;
  for (int j = 0; j < per; ++j) s += h[tid * per + j];
  ps[tid] = s;
  __syncthreads();
  if (tid == 0) {
    if (c == 0) {
      if (level == 3) med[t * NIMG + img] = 0.0f;
      else { bout[t * NIMG + img] = 0; rout[t * NIMG + img] = 0; }
    } else {
      unsigned cum = 0;
      int g = 0;
      for (; g < 255; ++g) {
        if (cum + ps[g] > k) break;
        cum += ps[g];
      }
      int b = g * per;
      for (int j = 0; j < per; ++j) {
        unsigned hv = h[g * per + j];
        if (cum + hv > k) { b = g * per + j; break; }
        cum += hv;
      }
      if (level == 3) {
        unsigned key = (b1[t * NIMG + img] << 21) |
                       (b2[t * NIMG + img] << 10) | (unsigned)b;
        med[t * NIMG + img] = fval(key);
      } else {
        bout[t * NIMG + img] = (unsigned)b;
        rout[t * NIMG + img] = k - cum;
      }
    }
  }
}

// ---- masked sum of |v - med| per (tensor, image) ----
__global__ __launch_bounds__(256) void k_abs(
    const float* __restrict__ pred, const float* __restrict__ yv,
    const u8* __restrict__ mask, const float* __restrict__ med,
    float* __restrict__ abs_part) {
  __shared__ float lds[256];
  const int img = blockIdx.y, t = blockIdx.z;
  const float* src = t ? yv : pred;
  const float mv = med[t * NIMG + img];
  const int base = img * NPIX;
  float s = 0.f;
  for (int i = blockIdx.x * 256 + threadIdx.x; i < NPIX; i += gridDim.x * 256)
    if (mask[base + i]) s += fabsf(src[base + i] - mv);
  float bs = block_sum(s, lds);
  if (threadIdx.x == 0)
    abs_part[(t * NIMG + img) * gridDim.x + blockIdx.x] = bs;
}

// ---- finalize scale: inv_sc = 1/(abs/cnt + eps); valid = max(cnt,1) ----
__global__ void k_scale(const float* __restrict__ abs_part,
                        const u32* __restrict__ cnt, float* __restrict__ inv,
                        float* __restrict__ valid) {
  const int tid = threadIdx.x;
  if (tid < 128) {
    const int img = tid & 63;
    float s = 0.f;
    for (int j = 0; j < 32; ++j) s += abs_part[tid * 32 + j];
    unsigned c = cnt[img];
    float sc = c ? (s / (float)c + EPSV) : EPSV;
    inv[tid] = 1.0f / sc;
    if (tid < 64) valid[tid] = (float)(c ? c : 1u);
  }
}

// ---- fused: rho sum per image + 2x2 pool to scale-1 diff/mask ----
__global__ __launch_bounds__(256) void k_pool1(
    const float* __restrict__ pred, const float* __restrict__ yv,
    const u8* __restrict__ mask, const float* __restrict__ med,
    const float* __restrict__ inv, float* __restrict__ diff1,
    u8* __restrict__ mask1, float* __restrict__ rho_part) {
  __shared__ float lds[256];
  const int img = blockIdx.y;
  const float mp = med[img], my = med[NIMG + img];
  const float ip = inv[img], iy = inv[NIMG + img];
  const int base = img * NPIX;
  float rho = 0.f;
  for (int c = blockIdx.x * 256 + threadIdx.x; c < C1; c += gridDim.x * 256) {
    int cy = c / W1, cx = c - cy * W1;
    int p0 = base + (cy * 2) * WD + cx * 2;
    float dsum = 0.f;
    int many = 0;
#pragma unroll
    for (int dy = 0; dy < 2; ++dy)
#pragma unroll
      for (int dx = 0; dx < 2; ++dx) {
        int pi = p0 + dy * WD + dx;
        float d = (pred[pi] - mp) * ip - (yv[pi] - my) * iy;
        dsum += d;
        u8 m = mask[pi];
        many |= m;
        rho += m ? fabsf(d) : 0.f;
      }
    diff1[img * C1 + c] = 0.25f * dsum;
    mask1[img * C1 + c] = many ? 1 : 0;
  }
  float rs = block_sum(rho, lds);
  if (threadIdx.x == 0) rho_part[img * gridDim.x + blockIdx.x] = rs;
}

// ---- scale-0 gradient sums (diff recomputed on the fly) ----
__global__ __launch_bounds__(256) void k_grad0(
    const float* __restrict__ pred, const float* __restrict__ yv,
    const u8* __restrict__ mask, const float* __restrict__ med,
    const float* __restrict__ inv, float* __restrict__ part) {
  __shared__ float lds[256];
  float gx = 0.f, cx = 0.f, gy = 0.f, cy = 0.f;
  const int stride = gridDim.x * 256;
  for (int idx = blockIdx.x * 256 + threadIdx.x; idx < TOT; idx += stride) {
    int img = idx / NPIX;
    int rem = idx - img * NPIX;
    int yy = rem / WD;
    int xx = rem - yy * WD;
    bool m0 = mask[idx] != 0;
    float mp = med[img], my = med[NIMG + img];
    float ip = inv[img], iy = inv[NIMG + img];
    float d0 = (pred[idx] - mp) * ip - (yv[idx] - my) * iy;
    if (xx + 1 < WD && m0 && mask[idx + 1]) {
      float d1 = (pred[idx + 1] - mp) * ip - (yv[idx + 1] - my) * iy;
      gx += fabsf(d0 - d1);
      cx += 1.f;
    }
    if (yy + 1 < HD && m0 && mask[idx + WD]) {
      float d1 = (pred[idx + WD] - mp) * ip - (yv[idx + WD] - my) * iy;
      gy += fabsf(d0 - d1);
      cy += 1.f;
    }
  }
  float s0 = block_sum(gx, lds), s1 = block_sum(cx, lds);
  float s2 = block_sum(gy, lds), s3 = block_sum(cy, lds);
  if (threadIdx.x == 0) {
    part[blockIdx.x * 4 + 0] = s0;
    part[blockIdx.x * 4 + 1] = s1;
    part[blockIdx.x * 4 + 2] = s2;
    part[blockIdx.x * 4 + 3] = s3;
  }
}

// ---- gradient sums on pooled arrays ----
__global__ __launch_bounds__(256) void k_grads(
    const float* __restrict__ diff, const u8* __restrict__ msk, int Hs, int Ws,
    float* __restrict__ part) {
  __shared__ float lds[256];
  const int n = Hs * Ws;
  const int tot = NIMG * n;
  float gx = 0.f, cx = 0.f, gy = 0.f, cy = 0.f;
  for (int idx = blockIdx.x * 256 + threadIdx.x; idx < tot;
       idx += gridDim.x * 256) {
    int img = idx / n;
    int rem = idx - img * n;
    int yy = rem / Ws;
    int xx = rem - yy * Ws;
    if (msk[idx]) {
      float d0 = diff[idx];
      if (xx + 1 < Ws && msk[idx + 1]) { gx += fabsf(d0 - diff[idx + 1]); cx += 1.f; }
      if (yy + 1 < Hs && msk[idx + Ws]) { gy += fabsf(d0 - diff[idx + Ws]); cy += 1.f; }
    }
  }
  float s0 = block_sum(gx, lds), s1 = block_sum(cx, lds);
  float s2 = block_sum(gy, lds), s3 = block_sum(cy, lds);
  if (threadIdx.x == 0) {
    part[blockIdx.x * 4 + 0] = s0;
    part[blockIdx.x * 4 + 1] = s1;
    part[blockIdx.x * 4 + 2] = s2;
    part[blockIdx.x * 4 + 3] = s3;
  }
}

// ---- pool-by-2 (exact nested mean == direct kxk mean; mask = any) ----
__global__ __launch_bounds__(256) void k_pool2(
    const float* __restrict__ din, const u8* __restrict__ min_, int Hi, int Wi,
    float* __restrict__ dout, u8* __restrict__ mout) {
  const int Ho = Hi >> 1, Wo = Wi >> 1;
  const int n = Ho * Wo;
  const int tot = NIMG * n;
  for (int idx = blockIdx.x * 256 + threadIdx.x; idx < tot;
       idx += gridDim.x * 256) {
    int img = idx / n;
    int rem = idx - img * n;
    int oy = rem / Wo;
    int ox = rem - oy * Wo;
    int b = img * Hi * Wi + (oy * 2) * Wi + ox * 2;
    float d = 0.25f * (din[b] + din[b + 1] + din[b + Wi] + din[b + Wi + 1]);
    u8 m = (u8)(min_[b] | min_[b + 1] | min_[b + Wi] | min_[b + Wi + 1]);
    dout[idx] = d;
    mout[idx] = m ? 1 : 0;
  }
}

// ---- final combine ----
__global__ __launch_bounds__(256) void k_final(
    const float* __restrict__ rho_part, const float* __restrict__ valid,
    const float* __restrict__ gpart, float* __restrict__ out) {
  __shared__ float lds[256];
  const int tid = threadIdx.x;
  float v = 0.f;
  if (tid < NIMG) {
    float r = 0.f;
    for (int j = 0; j < 64; ++j) r += rho_part[tid * 64 + j];
    v = r / valid[tid];
  }
  float ssi = block_sum(v, lds) * (1.0f / 64.0f);
  float gtot = 0.f;
  for (int s = 0; s < 4; ++s) {
    const float* gp = gpart + s * 4096;
    float a0 = 0.f, a1 = 0.f, a2 = 0.f, a3 = 0.f;
    for (int i = tid; i < 1024; i += 256) {
      a0 += gp[i * 4 + 0];
      a1 += gp[i * 4 + 1];
      a2 += gp[i * 4 + 2];
      a3 += gp[i * 4 + 3];
    }
    float gx = block_sum(a0, lds);
    float cx = block_sum(a1, lds);
    float gy = block_sum(a2, lds);
    float cy = block_sum(a3, lds);
    gtot += gx / fmaxf(cx, 1.f) + gy / fmaxf(cy, 1.f);
  }
  if (tid == 0) out[0] = ssi + 0.5f * (gtot * 0.25f);
}

extern "C" void kernel_launch(void* const* d_in, const int* in_sizes, int n_in,
                              void* d_out, int out_size, void* d_ws,
                              size_t ws_size, hipStream_t stream) {
  (void)in_sizes; (void)n_in; (void)out_size; (void)ws_size;
  const float* pred = (const float*)d_in[0];
  const float* yv = (const float*)d_in[1];
  const u8* mask = (const u8*)d_in[2];
  float* out = (float*)d_out;

  char* ws = (char*)d_ws;
  size_t off = 0;
  auto alloc = [&](size_t bytes) -> void* {
    void* p = ws + off;
    off = (off + bytes + 255) & ~(size_t)255;
    return p;
  };
  u32* hist1 = (u32*)alloc((size_t)2 * NIMG * 2048 * 4);
  u32* hist2 = (u32*)alloc((size_t)2 * NIMG * 2048 * 4);
  u32* hist3 = (u32*)alloc((size_t)2 * NIMG * 1024 * 4);
  u32* cnt = (u32*)alloc(NIMG * 4);
  const int zero_words = (int)(((char*)(cnt + NIMG) - (char*)hist1) / 4);
  u32* b1 = (u32*)alloc(128 * 4);
  u32* r1 = (u32*)alloc(128 * 4);
  u32* b2 = (u32*)alloc(128 * 4);
  u32* r2 = (u32*)alloc(128 * 4);
  float* med = (float*)alloc(128 * 4);
  float* inv = (float*)alloc(128 * 4);
  float* valid = (float*)alloc(NIMG * 4);
  float* abs_part = (float*)alloc((size_t)2 * NIMG * 32 * 4);
  float* rho_part = (float*)alloc((size_t)NIMG * 64 * 4);
  float* gpart = (float*)alloc((size_t)4 * 1024 * 4 * 4);
  float* diff1 = (float*)alloc((size_t)NIMG * C1 * 4);
  u8* mask1 = (u8*)alloc((size_t)NIMG * C1);
  float* diff2 = (float*)alloc((size_t)NIMG * C2 * 4);
  u8* mask2 = (u8*)alloc((size_t)NIMG * C2);
  float* diff3 = (float*)alloc((size_t)NIMG * C3 * 4);
  u8* mask3 = (u8*)alloc((size_t)NIMG * C3);

  // 0) zero histograms + counts (workspace is poisoned)
  k_zero<<<256, 256, 0, stream>>>(hist1, zero_words);

  // 1-3) exact masked lower-median via 3-level radix select (11/11/10 bits)
  dim3 hgrid(16, NIMG, 2);
  k_hist<<<hgrid, 256, 0, stream>>>(pred, yv, mask, hist1, cnt, b1, b2, 1, 2048);
  k_select<<<128, 256, 0, stream>>>(hist1, cnt, r1, b1, r1, b1, b2, med, 1, 2048);
  k_hist<<<hgrid, 256, 0, stream>>>(pred, yv, mask, hist2, cnt, b1, b2, 2, 2048);
  k_select<<<128, 256, 0, stream>>>(hist2, cnt, r1, b2, r2, b1, b2, med, 2, 2048);
  k_hist<<<hgrid, 256, 0, stream>>>(pred, yv, mask, hist3, cnt, b1, b2, 3, 1024);
  k_select<<<128, 256, 0, stream>>>(hist3, cnt, r2, b2, r2, b1, b2, med, 3, 1024);

  // 4) masked MAD scale
  k_abs<<<dim3(32, NIMG, 2), 256, 0, stream>>>(pred, yv, mask, med, abs_part);
  k_scale<<<1, 256, 0, stream>>>(abs_part, cnt, inv, valid);

  // 5) fused rho-sum + pool to scale 1
  k_pool1<<<dim3(64, NIMG), 256, 0, stream>>>(pred, yv, mask, med, inv, diff1,
                                              mask1, rho_part);

  // 6) multi-scale gradient loss
  k_grad0<<<1024, 256, 0, stream>>>(pred, yv, mask, med, inv, gpart + 0 * 4096);
  k_grads<<<1024, 256, 0, stream>>>(diff1, mask1, H1, W1, gpart + 1 * 4096);
  k_pool2<<<512, 256, 0, stream>>>(diff1, mask1, H1, W1, diff2, mask2);
  k_grads<<<1024, 256, 0, stream>>>(diff2, mask2, 129, 129, gpart + 2 * 4096);
  k_pool2<<<256, 256, 0, stream>>>(diff2, mask2, 129, 129, diff3, mask3);
  k_grads<<<1024, 256, 0, stream>>>(diff3, mask3, 64, 64, gpart + 3 * 4096);

  // 7) combine
  k_final<<<1, 256, 0, stream>>>(rho_part, valid, gpart, out);
}